// GCN_37495064494793
// MI455X (gfx1250) — compile-verified
//
#include <hip/hip_runtime.h>
#include <stdint.h>

#define D_IN  128
#define D_OUT 32

typedef float v2f __attribute__((ext_vector_type(2)));
typedef float v8f __attribute__((ext_vector_type(8)));

// ---------------------------------------------------------------------------
// Kernel 1: deg[i] = 1.0f  (self-loop contribution to target degree)
// ---------------------------------------------------------------------------
__global__ __launch_bounds__(256) void gcn_deg_init(float* __restrict__ deg, int n) {
  int i = blockIdx.x * 256 + threadIdx.x;
  if (i < n) deg[i] = 1.0f;
}

// ---------------------------------------------------------------------------
// Kernel 2: deg[col[e]] += 1 for real edges (L2-resident atomics)
// ---------------------------------------------------------------------------
__global__ __launch_bounds__(256) void gcn_deg_edges(const long long* __restrict__ col,
                                                     float* __restrict__ deg, int e) {
  int i = blockIdx.x * 256 + threadIdx.x;
  if (i < e) {
    long long c = col[i];
    atomicAdd(&deg[c], 1.0f);
  }
}

// ---------------------------------------------------------------------------
// Kernel 3: dis = rsqrt(deg)  (deg >= 1 always, so no zero branch needed,
// but keep the guard to match reference semantics exactly)
// ---------------------------------------------------------------------------
__global__ __launch_bounds__(256) void gcn_rsqrt(float* __restrict__ deg, int n) {
  int i = blockIdx.x * 256 + threadIdx.x;
  if (i < n) {
    float d = deg[i];
    deg[i] = (d > 0.0f) ? rsqrtf(d) : 0.0f;
  }
}

// ---------------------------------------------------------------------------
// Kernel 4: xw = x @ W via V_WMMA_F32_16X16X4_F32; also initialize
//   out[i][:] = dis[i]^2 * xw[i][:] + b[:]   (self-loop message + bias)
// One wave computes a 16-row x 32-col output tile (two 16x16 WMMA chains).
// ---------------------------------------------------------------------------
__global__ __launch_bounds__(256) void gcn_gemm_selfloop(
    const float* __restrict__ x,    // [n,128]
    const float* __restrict__ W,    // [128,32]
    const float* __restrict__ bias, // [32]
    const float* __restrict__ dis,  // [n]
    float* __restrict__ xw,         // [n,32]  scratch
    float* __restrict__ out,        // [n,32]
    int n)
{
  __shared__ float sW[D_IN * D_OUT]; // 16 KB
  int tid = threadIdx.x;
  for (int i = tid; i < D_IN * D_OUT; i += 256) sW[i] = W[i];
  __syncthreads();

  int wave = tid >> 5;
  int lane = tid & 31;
  int rowTile = blockIdx.x * 8 + wave;       // 16 rows per tile
  int rowBase = rowTile * 16;
  if (rowBase >= n) return;                  // wave-uniform: EXEC stays all-1s

  int m     = lane & 15;                     // row within tile (A) / column (B,C)
  int khalf = (lane >> 4) * 2;               // lanes 0-15: K=0,1 ; lanes 16-31: K=2,3

  int arow = rowBase + m;
  if (arow >= n) arow = n - 1;               // safe clamp (n % 16 == 0 here)
  const float* xr = x + (size_t)arow * D_IN;

  v8f acc0 = {};                             // columns 0..15
  v8f acc1 = {};                             // columns 16..31

  for (int k = 0; k < D_IN; k += 4) {
    // A fragment: x[rowBase+m][k+khalf .. k+khalf+1], 8B aligned
    v2f a = *(const v2f*)(xr + k + khalf);
    // B fragments: W[k+khalf..+1][m] and W[k+khalf..+1][16+m]
    v2f b0, b1;
    b0.x = sW[(k + khalf)     * D_OUT + m];
    b0.y = sW[(k + khalf + 1) * D_OUT + m];
    b1.x = sW[(k + khalf)     * D_OUT + 16 + m];
    b1.y = sW[(k + khalf + 1) * D_OUT + 16 + m];
    acc0 = __builtin_amdgcn_wmma_f32_16x16x4_f32(false, a, false, b0,
                                                 (short)0, acc0, false, false);
    acc1 = __builtin_amdgcn_wmma_f32_16x16x4_f32(false, a, false, b1,
                                                 (short)0, acc1, false, false);
  }

  // C/D layout: VGPR r -> M = r (lanes 0-15) or M = r+8 (lanes 16-31); N = lane&15
  int ncol = lane & 15;
  int mOff = (lane >> 4) * 8;
  float b_lo = bias[ncol];
  float b_hi = bias[16 + ncol];
  #pragma unroll
  for (int r = 0; r < 8; ++r) {
    int orow = rowBase + mOff + r;
    if (orow < n) {
      float s  = dis[orow];
      float ss = s * s;
      float v0 = acc0[r];
      float v1 = acc1[r];
      size_t base = (size_t)orow * D_OUT;
      xw [base + ncol]      = v0;
      xw [base + 16 + ncol] = v1;
      out[base + ncol]      = ss * v0 + b_lo;
      out[base + 16 + ncol] = ss * v1 + b_hi;
    }
  }
}

// ---------------------------------------------------------------------------
// Kernel 5: edge scatter. One wave per edge: lane = output channel.
// 128B coalesced gather of xw[src] (L2-resident) + 32 lane-parallel
// global_atomic_add_f32 into out[dst].
// ---------------------------------------------------------------------------
__global__ __launch_bounds__(256) void gcn_edge_scatter(
    const long long* __restrict__ ei,   // [2, e] flat: src at [0,e), dst at [e,2e)
    const float* __restrict__ dis,
    const float* __restrict__ xw,
    float* __restrict__ out,
    int e)
{
  int lane   = threadIdx.x & 31;
  int waveId = (blockIdx.x * 256 + threadIdx.x) >> 5;
  if (waveId >= e) return;
  long long src = ei[waveId];
  long long dst = ei[(size_t)e + waveId];
  float nrm = dis[src] * dis[dst];
  float v   = nrm * xw[(size_t)src * D_OUT + lane];
  atomicAdd(&out[(size_t)dst * D_OUT + lane], v);
}

// ---------------------------------------------------------------------------
extern "C" void kernel_launch(void* const* d_in, const int* in_sizes, int n_in,
                              void* d_out, int out_size, void* d_ws, size_t ws_size,
                              hipStream_t stream) {
  const float*     x  = (const float*)d_in[0];       // [n,128] fp32
  const long long* ei = (const long long*)d_in[1];   // [2,e] int64
  const float*     W  = (const float*)d_in[2];       // [128,32] fp32
  const float*     b  = (const float*)d_in[3];       // [32] fp32
  float* out = (float*)d_out;                        // [n,32] fp32

  int n = in_sizes[0] / D_IN;
  int e = in_sizes[1] / 2;

  // Workspace: [0, n) floats = deg/dis ; then n*32 floats = xw
  float* deg = (float*)d_ws;
  size_t off = (((size_t)n * sizeof(float)) + 255) & ~(size_t)255;
  float* xw  = (float*)((char*)d_ws + off);

  gcn_deg_init <<<(n + 255) / 256, 256, 0, stream>>>(deg, n);
  gcn_deg_edges<<<(e + 255) / 256, 256, 0, stream>>>(ei + e, deg, e);
  gcn_rsqrt    <<<(n + 255) / 256, 256, 0, stream>>>(deg, n);

  int tiles  = (n + 15) / 16;
  int blocks = (tiles + 7) / 8;     // 8 waves (tiles) per 256-thread block
  gcn_gemm_selfloop<<<blocks, 256, 0, stream>>>(x, W, b, deg, xw, out, n);

  long long threads = (long long)e * 32;
  gcn_edge_scatter<<<(int)((threads + 255) / 256), 256, 0, stream>>>(ei, deg, xw, out, e);
}